// TimeConcater_12266426597511
// MI455X (gfx1250) — compile-verified
//
#include <hip/hip_runtime.h>

#define NUM_TB 32      // time blocks (P)
#define Bsz    32
#define Lsz    4096
#define Dsz    256

typedef __attribute__((ext_vector_type(2))) float v2f;
typedef __attribute__((ext_vector_type(8))) float v8f;

// Grid: (8 d-tiles of 32 columns, 32 batches). Block: 512 threads = 16 waves.
// Wave w: dsub = w&1 (16 d columns), K-chunk c = w>>1 (512 rows of L).
// Each wave accumulates two 16x16 fp32 WMMA tiles (p=0..15, p=16..31) over its
// K chunk, then all 16 waves reduce through LDS and store the (32p x 32d) tile.
__global__ __launch_bounds__(512)
void time_bin_wmma_kernel(const float* __restrict__ x,
                          const float* __restrict__ ts,
                          float* __restrict__ out) {
    __shared__ unsigned int s_bins[Lsz / 4];   // 4 packed 8-bit bins per dword
    __shared__ float        s_red[16 * 512];   // per-wave partial tiles

    const int tid = threadIdx.x;
    const int b   = blockIdx.y;
    const int d0  = blockIdx.x * 32;

    // ---- passthrough output: copy time_steps (d-tile 0 workgroups only) ----
    if (blockIdx.x == 0) {
        const float* tsb = ts + b * Lsz;
        float* o2 = out + (size_t)Bsz * NUM_TB * Dsz + (size_t)b * Lsz;
#pragma unroll
        for (int i = 0; i < Lsz / 512; ++i)
            o2[tid + i * 512] = tsb[tid + i * 512];
    }

    // ---- bin indices for this batch, packed 4 per dword into LDS ----
    // bins p: [1.001*p/32, 1.001*(p+1)/32)  =>  p = floor(ts * 32/1.001)
    const float scale = 32.0f / 1.001f;
    {
        const float4* ts4 = (const float4*)(ts + b * Lsz);
#pragma unroll
        for (int g = tid; g < Lsz / 4; g += 512) {
            float4 t = ts4[g];
            int b0 = (int)(t.x * scale); b0 = b0 < 0 ? 0 : (b0 > 31 ? 31 : b0);
            int b1 = (int)(t.y * scale); b1 = b1 < 0 ? 0 : (b1 > 31 ? 31 : b1);
            int b2 = (int)(t.z * scale); b2 = b2 < 0 ? 0 : (b2 > 31 ? 31 : b2);
            int b3 = (int)(t.w * scale); b3 = b3 < 0 ? 0 : (b3 > 31 ? 31 : b3);
            s_bins[g] = (unsigned)b0 | ((unsigned)b1 << 8) |
                        ((unsigned)b2 << 16) | ((unsigned)b3 << 24);
        }
    }
    __syncthreads();

    const int wave = tid >> 5;
    const int lane = tid & 31;
    const int dsub = wave & 1;      // which 16 of the 32 d columns
    const int kc   = wave >> 1;     // K chunk 0..7 (512 rows each)
    const int h    = lane >> 4;     // lane half: selects K pair {2h, 2h+1}
    const int n    = lane & 15;     // A row (p) / B column (d) within tile

    // Per-lane streaming pointer into x: row l = kc*512 + 2h, col d0+dsub*16+n
    const float* xp = x + ((size_t)b * Lsz + (size_t)(kc * 512 + 2 * h)) * Dsz
                        + (d0 + dsub * 16 + n);

    v8f acc0 = {};   // p = 0..15
    v8f acc1 = {};   // p = 16..31
    const unsigned M1 = (unsigned)(n + 16);

    for (int i = 0; i < 128; ++i) {          // 128 steps of K=4 => 512 rows
        // packed bins for rows k0..k0+3 (wave-uniform LDS broadcast)
        unsigned packed = s_bins[kc * 128 + i];
        unsigned e0 = (packed >> (16 * h)) & 0xFFu;       // bin(l = k0+2h)
        unsigned e1 = (packed >> (16 * h + 8)) & 0xFFu;   // bin(l = k0+2h+1)

        v2f A0, A1, Bv;
        A0.x = (e0 == (unsigned)n) ? 1.0f : 0.0f;   // one-hot mask tile rows 0..15
        A0.y = (e1 == (unsigned)n) ? 1.0f : 0.0f;
        A1.x = (e0 == M1) ? 1.0f : 0.0f;            // rows 16..31
        A1.y = (e1 == M1) ? 1.0f : 0.0f;

        // B tile 4x16 fp32: lanes 0-15 K={k0,k0+1}, lanes 16-31 K={k0+2,k0+3}
        Bv.x = __builtin_nontemporal_load(xp);
        Bv.y = __builtin_nontemporal_load(xp + Dsz);

        acc0 = __builtin_amdgcn_wmma_f32_16x16x4_f32(
                   false, A0, false, Bv, (short)0, acc0, false, false);
        acc1 = __builtin_amdgcn_wmma_f32_16x16x4_f32(
                   false, A1, false, Bv, (short)0, acc1, false, false);

        xp += 4 * Dsz;
    }

    // ---- deterministic cross-wave reduction through LDS ----
#pragma unroll
    for (int r = 0; r < 8; ++r) {
        s_red[wave * 512 +       r * 32 + lane] = acc0[r];
        s_red[wave * 512 + 256 + r * 32 + lane] = acc1[r];
    }
    __syncthreads();

    {
        const int j  = tid;            // 512 threads x 2 outputs = 1024 elems
        const int t  = j >> 8;         // 0: p<16, 1: p>=16
        const int r  = (j >> 5) & 7;   // C/D VGPR index
        const int ln = j & 31;         // C/D lane
        const int p  = t * 16 + r + 8 * (ln >> 4);   // ISA 16x16 f32 C layout
        const int dd = ln & 15;

        float s0 = 0.0f, s1 = 0.0f;    // sum the 8 K-chunk partials per dsub
#pragma unroll
        for (int c = 0; c < 8; ++c) {
            s0 += s_red[(2 * c + 0) * 512 + j];
            s1 += s_red[(2 * c + 1) * 512 + j];
        }
        float* ob = out + ((size_t)b * NUM_TB + p) * Dsz + d0;
        ob[0 * 16 + dd] = s0;
        ob[1 * 16 + dd] = s1;
    }
}

extern "C" void kernel_launch(void* const* d_in, const int* in_sizes, int n_in,
                              void* d_out, int out_size, void* d_ws, size_t ws_size,
                              hipStream_t stream) {
    (void)in_sizes; (void)n_in; (void)d_ws; (void)ws_size; (void)out_size;
    const float* x  = (const float*)d_in[0];
    const float* ts = (const float*)d_in[1];
    float* out = (float*)d_out;

    dim3 grid(Dsz / 32, Bsz);   // (8 d-tiles, 32 batches)
    time_bin_wmma_kernel<<<grid, 512, 0, stream>>>(x, ts, out);
}